// Transformer_Reconstruction_Four_Channels_65103114272872
// MI455X (gfx1250) — compile-verified
//
#include <hip/hip_runtime.h>

// Problem constants (match reference)
#define KB 32
#define KN 400
#define KC 720
#define KH 12
#define KHD 60
#define KNT 25    // KN / 16
#define KSTR 68   // LDS row stride (floats): 16B-aligned rows, conflict-free b64 reads

typedef __attribute__((ext_vector_type(2))) float v2f;
typedef __attribute__((ext_vector_type(8))) float v8f;
typedef __attribute__((ext_vector_type(4))) int v4i;

__device__ __forceinline__ v8f wmma_f32_16x16x4(v2f a, v2f b, v8f c) {
  // (neg_a, A, neg_b, B, c_mod, C, reuse_a, reuse_b)
  return __builtin_amdgcn_wmma_f32_16x16x4_f32(false, a, false, b, (short)0, c,
                                               false, false);
}

// ---- CDNA5 async global->LDS copy (ASYNCcnt path), with sync fallback ------
#if defined(__AMDGCN__) && __has_builtin(__builtin_amdgcn_global_load_async_to_lds_b128)
#define HAVE_ASYNC_LDS 1
#else
#define HAVE_ASYNC_LDS 0
#endif

__device__ __forceinline__ void copy16_to_lds(float* lds, const float* g) {
#if HAVE_ASYNC_LDS
  // signature (from clang diagnostic): (v4i __device__*, v4i __shared__*, int, int)
  __builtin_amdgcn_global_load_async_to_lds_b128(
      (__attribute__((address_space(1))) v4i*)g,
      (__attribute__((address_space(3))) v4i*)lds, 0, 0);
#else
  *(float4*)lds = *(const float4*)g;
#endif
}

__device__ __forceinline__ void async_join() {
#if HAVE_ASYNC_LDS
#if __has_builtin(__builtin_amdgcn_s_wait_asynccnt)
  __builtin_amdgcn_s_wait_asynccnt(0);
#else
  asm volatile("s_wait_asynccnt 0" ::: "memory");
#endif
#endif
}

// monotone float <-> uint encoding so atomicMax(unsigned) orders like float
__device__ __forceinline__ unsigned fenc(float f) {
  unsigned u = __float_as_uint(f);
  return (u & 0x80000000u) ? ~u : (u | 0x80000000u);
}
__device__ __forceinline__ float fdec(unsigned u) {
  unsigned b = (u & 0x80000000u) ? (u ^ 0x80000000u) : ~u;
  return __uint_as_float(b);
}

// ---------------------------------------------------------------------------
// Kernel 1: roi_mask[b,n] = any_c(mask[b,c,n] != 0)
// ---------------------------------------------------------------------------
__global__ __launch_bounds__(256) void roi_mask_kernel(
    const int* __restrict__ mask, int* __restrict__ roi) {
  int g = blockIdx.x * 256 + threadIdx.x;
  if (g >= KB * KN) return;
  int b = g / KN, n = g % KN;
  const int* p = mask + (size_t)b * KC * KN + n;
  int acc = 0;
#pragma unroll 8
  for (int c = 0; c < KC; ++c) acc |= p[(size_t)c * KN];
  roi[g] = (acc != 0) ? 1 : 0;
}

// ---------------------------------------------------------------------------
// Kernel 2: Q/K projection.  Flattened rows M = B*N = 12800.
// qbuf[m, f] = sum_c x[m,c] * w[f,c]          (f in [0,720)  -> Q)
// kbuf[m, f] = sum_c x[m,c] * w[720+f, c]     (f in [0,720)  -> K)
// One wave per 16x48 output tile: 800 M-tiles x 30 feature-groups of 48.
// All operands L2-resident; loads pipeline against WMMA.
// ---------------------------------------------------------------------------
__global__ __launch_bounds__(256) void qk_proj_kernel(
    const float* __restrict__ x, const float* __restrict__ w,
    float* __restrict__ qbuf, float* __restrict__ kbuf) {
  const int wave = threadIdx.x >> 5;
  const int lane = threadIdx.x & 31;
  const int gw = blockIdx.x * 8 + wave;
  const int m0 = __builtin_amdgcn_readfirstlane((gw / 30) * 16);  // row tile
  const int f0 = __builtin_amdgcn_readfirstlane((gw % 30) * 48);  // feat group
  const int lm = lane & 15;
  const int half = lane >> 4;

  const float* arow = x + (size_t)(m0 + lm) * KC + 2 * half;
  const float* b0 = w + (size_t)(f0 + lm) * KC + 2 * half;
  const float* b1 = b0 + (size_t)16 * KC;
  const float* b2 = b0 + (size_t)32 * KC;

  v8f acc0 = {0.f, 0.f, 0.f, 0.f, 0.f, 0.f, 0.f, 0.f};
  v8f acc1 = acc0, acc2 = acc0;
  for (int c0 = 0; c0 < KC; c0 += 4) {
    v2f a = *(const v2f*)(arow + c0);
    v2f q0 = *(const v2f*)(b0 + c0);
    v2f q1 = *(const v2f*)(b1 + c0);
    v2f q2 = *(const v2f*)(b2 + c0);
    acc0 = wmma_f32_16x16x4(a, q0, acc0);
    acc1 = wmma_f32_16x16x4(a, q1, acc1);
    acc2 = wmma_f32_16x16x4(a, q2, acc2);
  }

  // D layout: VGPR r, lanes 0-15 -> M=r, lanes 16-31 -> M=r+8; N = lane%16
  float* dst = (f0 < KC) ? qbuf : kbuf;
  const int fbase = (f0 < KC) ? f0 : (f0 - KC);
#pragma unroll
  for (int r = 0; r < 8; ++r) {
    float* drow = dst + (size_t)(m0 + r + 8 * half) * KC + fbase + lm;
    drow[0] = acc0[r];
    drow[16] = acc1[r];
    drow[32] = acc2[r];
  }
}

// ---------------------------------------------------------------------------
// Kernel 3: attention scores + mask + softmax, one block per (b, h, i-tile).
// Q tile and K slice staged in LDS (async global->LDS), K in 4 chunks of
// 128 rows; chunk c gives wave w the j-tile (c*8 + w).  Fragment reads are
// ds_load_b64, bank-conflict-free at row stride 68.
// ---------------------------------------------------------------------------
__global__ __launch_bounds__(256) void attn_softmax_kernel(
    const float* __restrict__ qbuf, const float* __restrict__ kbuf,
    const int* __restrict__ roi, float* __restrict__ out) {
  extern __shared__ float smem[];
  float* qs = smem;               // 16  x KSTR
  float* ks = smem + 16 * KSTR;   // 128 x KSTR
  __shared__ int roi_s[KN];
  __shared__ unsigned rowmax_u[16];
  __shared__ float rowsum_s[16];

  const int it = blockIdx.x % KNT;
  const int bh = blockIdx.x / KNT;
  const int h = bh % KH;
  const int b = bh / KH;
  const int i0 = it * 16;

  const int tid = threadIdx.x;
  for (int t = tid; t < KN; t += 256) roi_s[t] = roi[b * KN + t];
  if (tid < 16) {
    rowmax_u[tid] = 0u;  // fdec(0) is below every encoded float
    rowsum_s[tid] = 0.0f;
  }

  // stage Q tile: 16 rows x 15 x b128 (rows 16B-aligned: all offsets %4==0)
  const float* qg = qbuf + (size_t)(b * KN + i0) * KC + h * KHD;
  if (tid < 240) {
    int r = tid / 15, cq = tid % 15;
    copy16_to_lds(&qs[r * KSTR + 4 * cq], qg + (size_t)r * KC + 4 * cq);
  }

  const int wave = __builtin_amdgcn_readfirstlane(tid >> 5);
  const int lane = tid & 31;
  const int lm = lane & 15;
  const int half = lane >> 4;
  const int ntile = __builtin_amdgcn_readfirstlane((32 - wave) >> 3);  // 4 or 3

  v8f acc[4];
#pragma unroll
  for (int t = 0; t < 4; ++t)
    acc[t] = (v8f){0.f, 0.f, 0.f, 0.f, 0.f, 0.f, 0.f, 0.f};

  const float* kg = kbuf + (size_t)(b * KN) * KC + h * KHD;
#pragma unroll
  for (int c = 0; c < 4; ++c) {
    const int r0 = c * 128;
    const int nrows = (c == 3) ? 16 : 128;
    // stage K chunk rows [r0, r0+nrows)
    for (int idx = tid; idx < nrows * 15; idx += 256) {
      int r = idx / 15, cq = idx % 15;
      copy16_to_lds(&ks[r * KSTR + 4 * cq],
                    kg + (size_t)(r0 + r) * KC + 4 * cq);
    }
    async_join();
    __syncthreads();

    if (c * 8 + wave < 25) {  // wave-uniform (sgpr) guard
      const float* brow = &ks[(wave * 16 + lm) * KSTR + 2 * half];
      const float* arow = &qs[lm * KSTR + 2 * half];
#pragma unroll
      for (int e0 = 0; e0 < KHD; e0 += 4) {
        v2f a = *(const v2f*)(arow + e0);
        v2f bb = *(const v2f*)(brow + e0);
        acc[c] = wmma_f32_16x16x4(a, bb, acc[c]);
      }
    }
    __syncthreads();  // chunk consumed before next overwrite
  }

  // scale + mask, per-lane row maxima (reg r -> row i0 + r + 8*half)
  const float scale = 0.12909944487358056f;  // 1/sqrt(60)
  float pmax[8];
#pragma unroll
  for (int r = 0; r < 8; ++r) pmax[r] = -3.0e38f;
#pragma unroll
  for (int t = 0; t < 4; ++t) {
    if (t < ntile) {
      const int j = (wave + 8 * t) * 16 + lm;
      const bool ok = (roi_s[j] != 0);
#pragma unroll
      for (int r = 0; r < 8; ++r) {
        float v = ok ? acc[t][r] * scale : -10000.0f;
        acc[t][r] = v;
        pmax[r] = fmaxf(pmax[r], v);
      }
    }
  }
  // reduce across the 16 lanes sharing a row set
#pragma unroll
  for (int r = 0; r < 8; ++r) {
    float m = pmax[r];
    m = fmaxf(m, __shfl_xor(m, 1, 32));
    m = fmaxf(m, __shfl_xor(m, 2, 32));
    m = fmaxf(m, __shfl_xor(m, 4, 32));
    m = fmaxf(m, __shfl_xor(m, 8, 32));
    pmax[r] = m;
  }
  if (lm == 0) {
#pragma unroll
    for (int r = 0; r < 8; ++r) atomicMax(&rowmax_u[r + 8 * half], fenc(pmax[r]));
  }
  __syncthreads();

  float rmax[8];
#pragma unroll
  for (int r = 0; r < 8; ++r) rmax[r] = fdec(rowmax_u[r + 8 * half]);

  float psum[8];
#pragma unroll
  for (int r = 0; r < 8; ++r) psum[r] = 0.0f;
#pragma unroll
  for (int t = 0; t < 4; ++t) {
    if (t < ntile) {
#pragma unroll
      for (int r = 0; r < 8; ++r) {
        float p = __expf(acc[t][r] - rmax[r]);
        acc[t][r] = p;
        psum[r] += p;
      }
    }
  }
#pragma unroll
  for (int r = 0; r < 8; ++r) {
    float s = psum[r];
    s += __shfl_xor(s, 1, 32);
    s += __shfl_xor(s, 2, 32);
    s += __shfl_xor(s, 4, 32);
    s += __shfl_xor(s, 8, 32);
    psum[r] = s;
  }
  if (lm == 0) {
#pragma unroll
    for (int r = 0; r < 8; ++r) atomicAdd(&rowsum_s[r + 8 * half], psum[r]);
  }
  __syncthreads();

  float rinv[8];
#pragma unroll
  for (int r = 0; r < 8; ++r) rinv[r] = 1.0f / rowsum_s[r + 8 * half];

  float* obase = out + ((size_t)(b * KH + h) * KN + i0) * KN;
#pragma unroll
  for (int t = 0; t < 4; ++t) {
    if (t < ntile) {
      const int j = (wave + 8 * t) * 16 + lm;
#pragma unroll
      for (int r = 0; r < 8; ++r)
        obase[(size_t)(r + 8 * half) * KN + j] = acc[t][r] * rinv[r];
    }
  }
}

// ---------------------------------------------------------------------------
extern "C" void kernel_launch(void* const* d_in, const int* in_sizes, int n_in,
                              void* d_out, int out_size, void* d_ws,
                              size_t ws_size, hipStream_t stream) {
  const float* x = (const float*)d_in[0];   // (B, N, C) fp32
  const int* mask = (const int*)d_in[1];    // (B, C, N) int32
  const float* w = (const float*)d_in[2];   // (3C, C) fp32
  float* out = (float*)d_out;               // (B, H, N, N) fp32

  // workspace: Qbuf | Kbuf | roi   (~73.8 MB, L2-resident)
  float* qbuf = (float*)d_ws;
  float* kbuf = qbuf + (size_t)KB * KN * KC;
  int* roi = (int*)(kbuf + (size_t)KB * KN * KC);

  roi_mask_kernel<<<(KB * KN + 255) / 256, 256, 0, stream>>>(mask, roi);
  qk_proj_kernel<<<3000, 256, 0, stream>>>(x, w, qbuf, kbuf);
  const size_t smem = (size_t)(16 + 128) * KSTR * sizeof(float);  // ~39 KB
  attn_softmax_kernel<<<KB * KH * KNT, 256, smem, stream>>>(qbuf, kbuf, roi,
                                                            out);
}